// MultiHeadSelfAttention_15040975471282
// MI455X (gfx1250) — compile-verified
//
#include <hip/hip_runtime.h>

typedef __attribute__((ext_vector_type(2))) float v2f;
typedef __attribute__((ext_vector_type(8))) float v8f;

#define WMMA_F32(a, b, c) \
  __builtin_amdgcn_wmma_f32_16x16x4_f32(false, (a), false, (b), (short)0, (c), false, false)

static constexpr int BB = 2, TT = 2048, CC = 1024, HH = 16, DD = 64;
static constexpr int MTOT = BB * TT;      // 4096 rows
static constexpr int NQKV = 3 * CC;       // 3072
static constexpr size_t HSZ = (size_t)BB * HH * TT * DD;  // 4194304 floats per tensor

// ---------------------------------------------------------------------------
// Kernel 1: QKV projection. One wave computes a 16x64 output tile of
// y = x @ W_qkv + b_qkv, and scatters it into [B,H,T,D]-layout q/k/v buffers.
// ---------------------------------------------------------------------------
__global__ __launch_bounds__(32)
void qkv_gemm_kernel(const float* __restrict__ X, const float* __restrict__ W,
                     const float* __restrict__ bias,
                     float* __restrict__ q, float* __restrict__ k,
                     float* __restrict__ v) {
  const int lane = threadIdx.x & 31;
  const int hl = lane >> 4;    // 0 or 1 (lane half)
  const int ln = lane & 15;
  const int n0 = blockIdx.x * 64;
  const int m0 = blockIdx.y * 16;

  v8f c[4] = {};
  const float* Arow = X + (size_t)(m0 + ln) * CC;   // A: M = ln

  for (int k0 = 0; k0 < CC; k0 += 4) {
    // A fragment: lane holds K = k0 + 2*hl + {0,1}  (contiguous -> b64 load)
    v2f a = *(const v2f*)(Arow + k0 + 2 * hl);
    __builtin_prefetch((const void*)(Arow + k0 + 64), 0, 0);
    const float* Bp = W + (size_t)(k0 + 2 * hl) * NQKV + n0 + ln;
#pragma unroll
    for (int n = 0; n < 4; ++n) {
      v2f b;
      b.x = Bp[16 * n];          // B row K = k0+2*hl,   N = n0+16n+ln
      b.y = Bp[NQKV + 16 * n];   // B row K = k0+2*hl+1
      c[n] = WMMA_F32(a, b, c[n]);
    }
  }

  // Epilogue: bias + scatter into q/k/v as [B,H,T,D]
#pragma unroll
  for (int n = 0; n < 4; ++n) {
    const int col = n0 + 16 * n + ln;        // 0..3071
    const float bval = bias[col];
    const int which = col >> 10;             // 0=q, 1=k, 2=v
    const int rem = col & (CC - 1);
    const int h = rem >> 6;
    const int d = rem & 63;
    float* dst = (which == 0) ? q : (which == 1) ? k : v;
#pragma unroll
    for (int r = 0; r < 8; ++r) {
      const int trow = m0 + r + 8 * hl;      // global token row
      const int bidx = trow >> 11;           // / 2048
      const int t = trow & (TT - 1);
      const size_t idx = (((size_t)(bidx * HH + h) * TT) + t) * DD + d;
      dst[idx] = c[n][r] + bval;
    }
  }
}

// ---------------------------------------------------------------------------
// Kernel 2: causal flash attention. One wave per (b,h, 16-query tile).
// ---------------------------------------------------------------------------
__global__ __launch_bounds__(32)
void attn_kernel(const float* __restrict__ q, const float* __restrict__ kmat,
                 const float* __restrict__ vmat, float* __restrict__ out) {
  const int lane = threadIdx.x & 31;
  const int hl = lane >> 4;
  const int ln = lane & 15;
  const int qtile = blockIdx.x;           // 0..127
  const int bh = blockIdx.y;              // 0..31
  const int b = bh >> 4, h = bh & 15;
  const int q0 = qtile * 16;

  const float* qbase = q + (size_t)bh * TT * DD;
  const float* kbase = kmat + (size_t)bh * TT * DD;
  const float* vbase = vmat + (size_t)bh * TT * DD;

  __shared__ float pt[16][17];            // P-tile transpose staging (padded)

  // Q A-fragments for all 16 K-steps over D=64, pre-scaled by 1/sqrt(D)=0.125
  v2f qa[16];
  {
    const float* qrow = qbase + (size_t)(q0 + ln) * DD + 2 * hl;
#pragma unroll
    for (int t = 0; t < 16; ++t) {
      v2f a = *(const v2f*)(qrow + 4 * t);
      qa[t] = a * 0.125f;
    }
  }

  v8f o[4] = {};
  float m_i[8], l_i[8];
#pragma unroll
  for (int r = 0; r < 8; ++r) { m_i[r] = -__builtin_inff(); l_i[r] = 0.0f; }

  for (int kt = 0; kt <= qtile; ++kt) {
    const int kb = kt * 16;

    // S = (Q/sqrt(D)) @ K^T : K is [T,D] row-major so the B^T fragment
    // (lane ln = key column, K-dim = d) is a contiguous float2 load.
    v8f s = {};
    const float* krow = kbase + (size_t)(kb + ln) * DD + 2 * hl;
#pragma unroll
    for (int t = 0; t < 16; ++t) {
      v2f bf = *(const v2f*)(krow + 4 * t);
      s = WMMA_F32(qa[t], bf, s);
    }

    const bool diag = (kt == qtile);
    float scale_r[8];
#pragma unroll
    for (int r = 0; r < 8; ++r) {
      const int row = r + 8 * hl;           // query row within tile
      float sv = s[r];
      if (diag && (ln > row)) sv = -__builtin_inff();   // causal mask
      // row max across the 16 lanes holding this row
      float mx = sv;
      mx = fmaxf(mx, __shfl_xor(mx, 1));
      mx = fmaxf(mx, __shfl_xor(mx, 2));
      mx = fmaxf(mx, __shfl_xor(mx, 4));
      mx = fmaxf(mx, __shfl_xor(mx, 8));
      const float mn = fmaxf(m_i[r], mx);
      const float sc = __expf(m_i[r] - mn);
      const float p = __expf(sv - mn);
      float ps = p;
      ps += __shfl_xor(ps, 1);
      ps += __shfl_xor(ps, 2);
      ps += __shfl_xor(ps, 4);
      ps += __shfl_xor(ps, 8);
      l_i[r] = l_i[r] * sc + ps;
      m_i[r] = mn;
      scale_r[r] = sc;
      pt[row][ln] = p;                       // C-layout -> LDS [qrow][key]
    }

    // rescale running O by exp(m_old - m_new) per row
#pragma unroll
    for (int n = 0; n < 4; ++n)
#pragma unroll
      for (int r = 0; r < 8; ++r) o[n][r] *= scale_r[r];

    __syncthreads();

    // O += P @ V : A = P (16q x 16k) from LDS, B = V tile (16k x 64d)
#pragma unroll
    for (int t = 0; t < 4; ++t) {
      v2f a;
      a.x = pt[ln][4 * t + 2 * hl];
      a.y = pt[ln][4 * t + 2 * hl + 1];
      const float* vrow = vbase + (size_t)(kb + 4 * t + 2 * hl) * DD + ln;
#pragma unroll
      for (int n = 0; n < 4; ++n) {
        v2f bf;
        bf.x = vrow[16 * n];        // key = kb+4t+2hl,   d = 16n+ln
        bf.y = vrow[DD + 16 * n];   // key = kb+4t+2hl+1
        o[n] = WMMA_F32(a, bf, o[n]);
      }
    }
    __syncthreads();
  }

  // Normalize and write to [B,T,C] for the output projection
#pragma unroll
  for (int r = 0; r < 8; ++r) {
    const float inv = 1.0f / l_i[r];
    const int trow = q0 + r + 8 * hl;
    float* orow = out + ((size_t)(b * TT + trow) * CC) + h * DD;
#pragma unroll
    for (int n = 0; n < 4; ++n) orow[16 * n + ln] = o[n][r] * inv;
  }
}

// ---------------------------------------------------------------------------
// Kernel 3: output projection y = A @ W_out + b_out, 16x64 tile per wave.
// ---------------------------------------------------------------------------
__global__ __launch_bounds__(32)
void out_gemm_kernel(const float* __restrict__ A, const float* __restrict__ W,
                     const float* __restrict__ bias, float* __restrict__ out) {
  const int lane = threadIdx.x & 31;
  const int hl = lane >> 4;
  const int ln = lane & 15;
  const int n0 = blockIdx.x * 64;
  const int m0 = blockIdx.y * 16;

  v8f c[4] = {};
  const float* Arow = A + (size_t)(m0 + ln) * CC;

  for (int k0 = 0; k0 < CC; k0 += 4) {
    v2f a = *(const v2f*)(Arow + k0 + 2 * hl);
    __builtin_prefetch((const void*)(Arow + k0 + 64), 0, 0);
    const float* Bp = W + (size_t)(k0 + 2 * hl) * CC + n0 + ln;
#pragma unroll
    for (int n = 0; n < 4; ++n) {
      v2f b;
      b.x = Bp[16 * n];
      b.y = Bp[CC + 16 * n];
      c[n] = WMMA_F32(a, b, c[n]);
    }
  }

#pragma unroll
  for (int n = 0; n < 4; ++n) {
    const float bval = bias[n0 + 16 * n + ln];
#pragma unroll
    for (int r = 0; r < 8; ++r) {
      out[(size_t)(m0 + r + 8 * hl) * CC + n0 + 16 * n + ln] = c[n][r] + bval;
    }
  }
}

// ---------------------------------------------------------------------------
extern "C" void kernel_launch(void* const* d_in, const int* in_sizes, int n_in,
                              void* d_out, int out_size, void* d_ws, size_t ws_size,
                              hipStream_t stream) {
  (void)in_sizes; (void)n_in; (void)out_size; (void)ws_size;
  const float* x    = (const float*)d_in[0];   // [2,2048,1024]
  const float* Wqkv = (const float*)d_in[1];   // [1024,3072]
  const float* bqkv = (const float*)d_in[2];   // [3072]
  const float* Wout = (const float*)d_in[3];   // [1024,1024]
  const float* bout = (const float*)d_in[4];   // [1024]
  float* out = (float*)d_out;                  // [2,2048,1024]

  float* qp = (float*)d_ws;        // [B,H,T,D]
  float* kp = qp + HSZ;
  float* vp = kp + HSZ;
  float* ao = vp + HSZ;            // attention output [B,T,C]

  const dim3 blk(32, 1, 1);
  qkv_gemm_kernel<<<dim3(NQKV / 64, MTOT / 16), blk, 0, stream>>>(x, Wqkv, bqkv, qp, kp, vp);
  attn_kernel<<<dim3(TT / 16, BB * HH), blk, 0, stream>>>(qp, kp, vp, ao);
  out_gemm_kernel<<<dim3(CC / 64, MTOT / 16), blk, 0, stream>>>(ao, Wout, bout, out);
}